// DELAY_72121090835120
// MI455X (gfx1250) — compile-verified
//
#include <hip/hip_runtime.h>
#include <hip/hip_bf16.h>

// Problem constants (from setup_inputs): x[B=32, T=2048, N=512], delays[N] in [0,64].
#define B_    32
#define T_    2048
#define N_    512
#define DMAX  64
#define TP    (T_ + DMAX)      // 2112 output time steps
#define TN    64               // channels per block tile
#define TT    192              // output time rows per block tile (2112 = 11 * 192)
#define ROWS  (TT + DMAX)      // 256 staged input rows (with halo)
#define NTHREADS 256           // 8 waves (wave32)

typedef float f4  __attribute__((ext_vector_type(4)));
typedef int   v4i __attribute__((ext_vector_type(4)));

#if defined(__gfx1250__) && __has_builtin(__builtin_amdgcn_global_load_async_to_lds_b128)
#define HAVE_ASYNC_LDS 1
typedef __attribute__((address_space(1))) v4i* gas_v4i_ptr;   // prints as 'v4i __device__ *'
typedef __attribute__((address_space(3))) v4i* lds_v4i_ptr;   // prints as 'v4i __shared__ *'
#else
#define HAVE_ASYNC_LDS 0
#endif

__global__ __launch_bounds__(NTHREADS)
void DELAY_tile_kernel(const float* __restrict__ x,
                       const int*   __restrict__ delays,
                       float*       __restrict__ out) {
    // 64 KiB LDS panel: ROWS x TN floats, row pitch 64 words -> bank = n mod 64
    __shared__ float tile[ROWS * TN];

    const int n0  = blockIdx.x * TN;   // channel tile base
    const int t0  = blockIdx.y * TT;   // output time tile base
    const int b   = blockIdx.z;        // batch
    const int tid = threadIdx.x;

    // ---------------- Stage-in: rows [t0-DMAX, t0+TT) x channels [n0, n0+TN) ----------------
    // 256 rows * 16 float4/row = 4096 float4 transfers; 16 per thread, coalesced.
    const int xbase = b * (T_ * N_);
    #pragma unroll
    for (int k = 0; k < (ROWS * (TN / 4)) / NTHREADS; ++k) {
        const int idx  = k * NTHREADS + tid;
        const int r    = idx >> 4;          // tile row
        const int q    = idx & 15;          // float4 column within row
        const int grow = t0 - DMAX + r;     // global time row (may be out of range)
        float* lp = &tile[r * TN + q * 4];
        if (grow >= 0 && grow < T_) {
            const float* gp = x + xbase + grow * N_ + n0 + q * 4;
#if HAVE_ASYNC_LDS
            // CDNA5 async DMA: global -> LDS, 16B per lane, tracked by ASYNCcnt.
            __builtin_amdgcn_global_load_async_to_lds_b128(
                (gas_v4i_ptr)gp, (lds_v4i_ptr)lp, /*offset=*/0, /*cpol=*/0);
#else
            *(f4*)lp = *(const f4*)gp;
#endif
        } else {
            f4 z = {0.f, 0.f, 0.f, 0.f};   // halo outside [0,T): zero-fill => free OOB logic
            *(f4*)lp = z;
        }
    }

#if HAVE_ASYNC_LDS
#if __has_builtin(__builtin_amdgcn_s_wait_asynccnt)
    __builtin_amdgcn_s_wait_asynccnt(0);
#else
    asm volatile("s_wait_asynccnt 0" ::: "memory");
#endif
#endif
    __syncthreads();

    // ---------------- Gather + store: out[b, t, n] = tile[(t - d[n]) - (t0 - DMAX)][n] --------
    const int tx = tid & (TN - 1);   // channel within tile (lane-contiguous -> conflict-free LDS)
    const int ty = tid >> 6;         // 0..3 time-phase
    int d = delays[n0 + tx];
    d = d < 0 ? 0 : (d > DMAX ? DMAX : d);

    const int obase = b * (TP * N_) + n0 + tx;
    #pragma unroll 4
    for (int toff = ty; toff < TT; toff += 4) {
        const int r = toff + DMAX - d;           // always in [0, ROWS)
        const float v = tile[r * TN + tx];
        // 132 MB streaming write: non-temporal so it doesn't evict L2-resident input.
        __builtin_nontemporal_store(v, out + obase + (t0 + toff) * N_);
    }
}

extern "C" void kernel_launch(void* const* d_in, const int* in_sizes, int n_in,
                              void* d_out, int out_size, void* d_ws, size_t ws_size,
                              hipStream_t stream) {
    const float* x      = (const float*)d_in[0];
    const int*   delays = (const int*)d_in[1];   // bridge doc: integer -> const int*
    float*       out    = (float*)d_out;
    (void)in_sizes; (void)n_in; (void)out_size; (void)d_ws; (void)ws_size;

    dim3 grid(N_ / TN, TP / TT, B_);   // (8, 11, 32) = 2816 blocks
    DELAY_tile_kernel<<<grid, NTHREADS, 0, stream>>>(x, delays, out);
}